// EmbNet_4741643895588
// MI455X (gfx1250) — compile-verified
//
#include <hip/hip_runtime.h>
#include <hip/hip_bf16.h>

typedef float v2f __attribute__((ext_vector_type(2)));
typedef float v8f __attribute__((ext_vector_type(8)));

#define UNITS 32
#define BN_EPS 1e-5f

__device__ __forceinline__ float sigm_f(float v) { return 1.0f / (1.0f + __expf(-v)); }
__device__ __forceinline__ float silu_f(float v) { return v * sigm_f(v); }

// ---------------------------------------------------------------------------
// Generic 32-channel GEMM via V_WMMA_F32_16X16X4_F32:
//   out[mat][m, n] = sum_k in[m,k] * W[mat][n,k] + B[mat][n]
// Each wave is pinned to ONE weight matrix (mat = wave % nmat); all 16
// B fragments (32 floats/lane) and both biases live in registers. The
// steady-state loop per 16-row tile:
//   8 coalesced b64 A-loads -> 2 interleaved chains of 8 fp32 WMMAs ->
//   16 stores via one base address + immediate offsets.
// K=32 reduced as 8 chained K=4 WMMAs in fp32 (precision == reference).
// M (rows) must be a multiple of 16 (N=100000 and E=1600000 both are).
// Requires (waves per block) % nmat == 0 (8 waves/block, nmat in {1,4}).
// ---------------------------------------------------------------------------
struct Gemm4 {
    const float* W[4];   // [32,32] row-major (out_ch, in_ch)
    const float* B[4];   // [32]
    float*       O[4];   // [M,32]
};

__global__ __launch_bounds__(256)
void gemm32_wmma(const float* __restrict__ in, Gemm4 g, int nmat, int ntiles) {
    const int lane = threadIdx.x & 31;
    const int rl   = lane & 15;        // M (A) or N (B/C) within tile
    const int half = lane >> 4;        // K-pair select
    const int wave = blockIdx.x * (blockDim.x >> 5) + (threadIdx.x >> 5);
    const int nwav = gridDim.x * (blockDim.x >> 5);

    const int mat        = wave % nmat;      // fixed matrix for this wave
    const int tile0      = wave / nmat;
    const int tileStride = nwav / nmat;

    // Hoist weights: B[k][n] = W[n][k]; lane rl = column n, half picks K pair.
    const float* __restrict__ Wm = g.W[mat];
    v2f bfrag[2][8];
    #pragma unroll
    for (int nt = 0; nt < 2; ++nt) {
        const float* wrow = Wm + (size_t)(nt * 16 + rl) * UNITS + 2 * half;
        #pragma unroll
        for (int s = 0; s < 8; ++s) {
            bfrag[nt][s].x = wrow[4 * s + 0];
            bfrag[nt][s].y = wrow[4 * s + 1];
        }
    }
    const float bias[2] = { g.B[mat][rl], g.B[mat][16 + rl] };
    float* __restrict__ Om = g.O[mat];

    const size_t aStep = (size_t)tileStride * 16 * UNITS;

    for (int tile = tile0; tile < ntiles; tile += tileStride) {
        // A fragments for 8 K-steps: lane rl holds row, half picks K pair.
        const float* arow = in + (size_t)(tile * 16 + rl) * UNITS + 2 * half;
        v2f a[8];
        #pragma unroll
        for (int s = 0; s < 8; ++s) {
            a[s].x = arow[4 * s + 0];
            a[s].y = arow[4 * s + 1];
        }
        // Speculative prefetch of next tile's A row (global_prefetch_b8):
        // OOB prefetches are silently dropped, so no guard / EXEC churn.
        __builtin_prefetch(arow + aStep, 0, 1);

        // One store base per ntile; consecutive C rows are 128B apart so the
        // per-v offsets fold into the instruction's immediate offset field.
        float* obase = Om + (size_t)(tile * 16 + 8 * half) * UNITS;

        #pragma unroll
        for (int nt = 0; nt < 2; ++nt) {
            v8f c;
            #pragma unroll
            for (int v = 0; v < 8; ++v) c[v] = bias[nt];  // bias per column n

            #pragma unroll
            for (int s = 0; s < 8; ++s) {
                c = __builtin_amdgcn_wmma_f32_16x16x4_f32(
                        false, a[s], false, bfrag[nt][s], (short)0, c, false, false);
            }
            // C layout: VGPR v -> M = v + 8*half, N = rl
            float* orow = obase + nt * 16 + rl;
            #pragma unroll
            for (int v = 0; v < 8; ++v) {
                orow[v * UNITS] = c[v];
            }
        }
    }
}

// ---------------------------------------------------------------------------
// Initial embeddings: x = silu(in @ W0.T + b0), F = 1 or 3 input features.
// Thread layout: lane = channel, wave = row (coalesced 128B rows).
// ---------------------------------------------------------------------------
__global__ __launch_bounds__(256)
void init_embed(const float* __restrict__ in, int F,
                const float* __restrict__ W, const float* __restrict__ B,
                float* __restrict__ out, int M) {
    const int c = threadIdx.x & 31;
    long r = (long)blockIdx.x * (blockDim.x >> 5) + (threadIdx.x >> 5);
    const long stride = (long)gridDim.x * (blockDim.x >> 5);
    for (; r < M; r += stride) {
        float acc = B[c];
        for (int f = 0; f < F; ++f)
            acc += in[r * F + f] * W[c * F + f];
        out[r * UNITS + c] = silu_f(acc);
    }
}

// ---------------------------------------------------------------------------
// Edge pass: msg = sigmoid(w) * x2[gat]; scatter-add into agg_sum/cnt;
// w_pre += x3[agg] + x4[gat]; accumulate edge-BN sum/sumsq per channel.
// Gathers hit L2 (x tensors are 12.8MB each). lane = channel, wave = edge.
// ---------------------------------------------------------------------------
__global__ __launch_bounds__(256)
void edge_msg(const float* __restrict__ w, float* __restrict__ wpre,
              const float* __restrict__ x2, const float* __restrict__ x3,
              const float* __restrict__ x4,
              const int* __restrict__ agg_idx, const int* __restrict__ gat_idx,
              float* __restrict__ agg_sum, float* __restrict__ cnt,
              float* __restrict__ estats, long E) {
    const int c = threadIdx.x & 31;
    long e = (long)blockIdx.x * (blockDim.x >> 5) + (threadIdx.x >> 5);
    const long stride = (long)gridDim.x * (blockDim.x >> 5);
    float s1 = 0.0f, s2 = 0.0f;
    for (; e < E; e += stride) {
        const int a = agg_idx[e];
        const int gi = gat_idx[e];
        const float wv = w[e * UNITS + c];
        const float m = sigm_f(wv) * x2[(size_t)gi * UNITS + c];
        atomicAdd(&agg_sum[(size_t)a * UNITS + c], m);
        if (c == 0) atomicAdd(&cnt[a], 1.0f);
        float pre = wpre[e * UNITS + c]
                  + x3[(size_t)a * UNITS + c]
                  + x4[(size_t)gi * UNITS + c];
        wpre[e * UNITS + c] = pre;
        s1 += pre;
        s2 += pre * pre;
    }
    atomicAdd(&estats[c], s1);
    atomicAdd(&estats[UNITS + c], s2);
}

// ---------------------------------------------------------------------------
// Node finalize: pre = x1 + agg_sum / max(cnt,1); node-BN stats.
// ---------------------------------------------------------------------------
__global__ __launch_bounds__(256)
void node_pre(float* __restrict__ x1, const float* __restrict__ agg_sum,
              const float* __restrict__ cnt, float* __restrict__ nstats, long M) {
    const int c = threadIdx.x & 31;
    long r = (long)blockIdx.x * (blockDim.x >> 5) + (threadIdx.x >> 5);
    const long stride = (long)gridDim.x * (blockDim.x >> 5);
    float s1 = 0.0f, s2 = 0.0f;
    for (; r < M; r += stride) {
        const float cn = fmaxf(cnt[r], 1.0f);
        float pre = x1[r * UNITS + c] + agg_sum[r * UNITS + c] / cn;
        x1[r * UNITS + c] = pre;
        s1 += pre;
        s2 += pre * pre;
    }
    atomicAdd(&nstats[c], s1);
    atomicAdd(&nstats[UNITS + c], s2);
}

// ---------------------------------------------------------------------------
// Apply: state += silu(BN(pre)); training-mode BN from global sum/sumsq.
// ---------------------------------------------------------------------------
__global__ __launch_bounds__(256)
void apply_bn(float* __restrict__ state, const float* __restrict__ pre,
              const float* __restrict__ stats, const float* __restrict__ gg,
              const float* __restrict__ bb, float invM, long M) {
    const int c = threadIdx.x & 31;
    const float mean = stats[c] * invM;
    const float var  = stats[UNITS + c] * invM - mean * mean;
    const float scale = rsqrtf(var + BN_EPS) * gg[c];
    const float shift = bb[c];
    long r = (long)blockIdx.x * (blockDim.x >> 5) + (threadIdx.x >> 5);
    const long stride = (long)gridDim.x * (blockDim.x >> 5);
    for (; r < M; r += stride) {
        const float p = pre[r * UNITS + c];
        state[r * UNITS + c] += silu_f((p - mean) * scale + shift);
    }
}

// ---------------------------------------------------------------------------
// Host-side orchestration.
// Input order (flattened setup_inputs dict):
//  0:x 1:y 2:edge_index 3:edge_attr, then 18 params per branch:
//  v0_w v0_b e0_w e0_b v1_w v1_b v2_w v2_b v3_w v3_b v4_w v4_b
//  el_w el_b vbn_g vbn_b ebn_g ebn_b
// ---------------------------------------------------------------------------
extern "C" void kernel_launch(void* const* d_in, const int* in_sizes, int n_in,
                              void* d_out, int out_size, void* d_ws, size_t ws_size,
                              hipStream_t stream) {
    const long N = in_sizes[0];                 // x: (N,1)
    const long E = in_sizes[3];                 // edge_attr: (E,1)
    const int depth = in_sizes[4 + 4] / (UNITS * UNITS);   // v1_w: (D,32,32)

    const int* ei = (const int*)d_in[2];
    const int* agg_idx = ei;
    const int* gat_idx = ei + E;
    const float* eattr = (const float*)d_in[3];

    // Workspace layout (floats)
    float* ws = (float*)d_ws;
    float* x       = ws;                         // N*32
    float* x1      = x  + N * UNITS;             // N*32
    float* x2      = x1 + N * UNITS;             // N*32
    float* x3      = x2 + N * UNITS;             // N*32
    float* x4      = x3 + N * UNITS;             // N*32
    float* agg_sum = x4 + N * UNITS;             // N*32
    float* cnt     = agg_sum + N * UNITS;        // N
    float* stats   = cnt + N;                    // 128: [node sum|sq][edge sum|sq]
    float* wpre    = stats + 128;                // E*32
    (void)ws_size; (void)n_in; (void)out_size;

    const int TB = 256;
    const int gNode = 512;    // node-sized row kernels (3.2M lane tasks)
    const int gEdge = 2048;   // edge-sized row kernels (51.2M lane tasks)
    const int gGemmN = 1024;  // 8192 waves -> 2048 waves per matrix
    const int gGemmE = 2048;  // 16384 waves on 100000 tiles

    for (int br = 0; br < 2; ++br) {
        void* const* P = d_in + 4 + br * 18;
        const float* feats = (const float*)d_in[br];       // x or y
        const int F = (br == 0) ? 1 : 3;
        float* w = (float*)d_out + (size_t)br * E * UNITS; // branch output, in-place state

        init_embed<<<gNode, TB, 0, stream>>>(feats, F, (const float*)P[0],
                                             (const float*)P[1], x, (int)N);
        init_embed<<<gEdge, TB, 0, stream>>>(eattr, 1, (const float*)P[2],
                                             (const float*)P[3], w, (int)E);

        for (int i = 0; i < depth; ++i) {
            hipMemsetAsync(agg_sum, 0, (size_t)N * UNITS * sizeof(float), stream);
            hipMemsetAsync(cnt, 0, (size_t)N * sizeof(float), stream);
            hipMemsetAsync(stats, 0, 128 * sizeof(float), stream);

            // x1..x4 = x @ v{1..4}_w[i].T + v{1..4}_b[i]
            Gemm4 gn;
            gn.W[0] = (const float*)P[4]  + (size_t)i * 1024;
            gn.W[1] = (const float*)P[6]  + (size_t)i * 1024;
            gn.W[2] = (const float*)P[8]  + (size_t)i * 1024;
            gn.W[3] = (const float*)P[10] + (size_t)i * 1024;
            gn.B[0] = (const float*)P[5]  + (size_t)i * UNITS;
            gn.B[1] = (const float*)P[7]  + (size_t)i * UNITS;
            gn.B[2] = (const float*)P[9]  + (size_t)i * UNITS;
            gn.B[3] = (const float*)P[11] + (size_t)i * UNITS;
            gn.O[0] = x1; gn.O[1] = x2; gn.O[2] = x3; gn.O[3] = x4;
            gemm32_wmma<<<gGemmN, TB, 0, stream>>>(x, gn, 4, (int)(N / 16));

            // w_pre = w @ el_w[i].T + el_b[i]
            Gemm4 ge;
            ge.W[0] = (const float*)P[12] + (size_t)i * 1024;
            ge.B[0] = (const float*)P[13] + (size_t)i * UNITS;
            ge.O[0] = wpre;
            ge.W[1] = ge.W[2] = ge.W[3] = ge.W[0];
            ge.B[1] = ge.B[2] = ge.B[3] = ge.B[0];
            ge.O[1] = ge.O[2] = ge.O[3] = wpre;
            gemm32_wmma<<<gGemmE, TB, 0, stream>>>(w, ge, 1, (int)(E / 16));

            // message/gather/scatter + edge BN stats
            edge_msg<<<gEdge, TB, 0, stream>>>(w, wpre, x2, x3, x4,
                                               agg_idx, gat_idx,
                                               agg_sum, cnt, stats + 64, E);

            // node pre-activation + node BN stats
            node_pre<<<gNode, TB, 0, stream>>>(x1, agg_sum, cnt, stats, N);

            // x += silu(BN(x1+agg)); w += silu(BN(w_pre))
            apply_bn<<<gNode, TB, 0, stream>>>(x, x1, stats,
                                               (const float*)P[14] + (size_t)i * UNITS,
                                               (const float*)P[15] + (size_t)i * UNITS,
                                               1.0f / (float)N, N);
            apply_bn<<<gEdge, TB, 0, stream>>>(w, wpre, stats + 64,
                                               (const float*)P[16] + (size_t)i * UNITS,
                                               (const float*)P[17] + (size_t)i * UNITS,
                                               1.0f / (float)E, E);
        }
    }
}